// UMPNN_64252710748275
// MI455X (gfx1250) — compile-verified
//
#include <hip/hip_runtime.h>

typedef float v2f __attribute__((ext_vector_type(2)));
typedef float v8f __attribute__((ext_vector_type(8)));

#define NNODES 4096
#define NEDGES 65536
#define IDIM   16
#define HDIM   16
#define KDIM   128
#define LN_EPS 1e-5f

__device__ __forceinline__ float sigmoidf_(float v) {
    return 1.0f / (1.0f + __expf(-v));
}

// Non-temporal loads for single-use weight streams (675 MB total > 192 MB L2):
// keep L2 for x / agg which have real reuse.
__device__ __forceinline__ v2f   nt_load2(const float* p) {
    return __builtin_nontemporal_load((const v2f*)p);
}
__device__ __forceinline__ float nt_load1(const float* p) {
    return __builtin_nontemporal_load(p);
}

__global__ void UMPNN_zero_kernel(float* __restrict__ p, int n) {
    int i = blockIdx.x * 256 + threadIdx.x;
    if (i < n) p[i] = 0.0f;
}

// ---------------------------------------------------------------------------
// Edge kernel: one wave (32 lanes) per edge.
// gates[64] = [W_ih | W_hh](64x32) @ concat(x[src], h0) via WMMA f32 16x16x4,
// B = z-chunk broadcast over all 16 columns (doubles as free result bcast).
// ---------------------------------------------------------------------------
__global__ __launch_bounds__(256) void UMPNN_edge_kernel(
    const float* __restrict__ x, const int* __restrict__ edge_index,
    const float* __restrict__ h0, const float* __restrict__ c0,
    const float* __restrict__ W_ih, const float* __restrict__ W_hh,
    const float* __restrict__ b_ih, const float* __restrict__ b_hh,
    float* __restrict__ agg)
{
    __shared__ __align__(16) float lds[8 * 96];   // per wave: 32 z + 64 gates
    const int lane = threadIdx.x & 31;
    const int wave = threadIdx.x >> 5;
    const int e = blockIdx.x * 8 + wave;

    float* zbuf = &lds[wave * 96];
    float* gbuf = zbuf + 32;

    const int tgt = edge_index[e];           // edge_index[0][e]
    const int src = edge_index[NEDGES + e];  // edge_index[1][e]

    // stage z = concat(x[src], h0) into per-wave LDS
    if (lane < 16) zbuf[lane] = x[src * IDIM + lane];
    else           zbuf[lane] = h0[lane - 16];
    __syncthreads();

    const float* Wih = W_ih + (size_t)e * 64 * 16;
    const float* Whh = W_hh + (size_t)e * 64 * 16;
    const int row = lane & 15;
    const int h2  = (lane >> 4) << 1;   // 0 (lanes 0-15) or 2 (lanes 16-31)

    v8f acc[4];
    #pragma unroll
    for (int rc = 0; rc < 4; ++rc) acc[rc] = (v8f){0,0,0,0,0,0,0,0};

    #pragma unroll
    for (int kc = 0; kc < 8; ++kc) {
        const int kk = kc * 4 + h2;                  // global K index (even)
        v2f b = *(const v2f*)&zbuf[kk];              // broadcast over N columns
        const float* Wbase = (kk < 16) ? (Wih + kk) : (Whh + (kk - 16));
        #pragma unroll
        for (int rc = 0; rc < 4; ++rc) {
            v2f a = nt_load2(Wbase + (rc * 16 + row) * 16);
            acc[rc] = __builtin_amdgcn_wmma_f32_16x16x4_f32(
                false, a, false, b, (short)0, acc[rc], false, false);
        }
    }

    // C layout: lanes 0-15 hold rows m..m+7, lanes 16-31 rows m+8..m+15
    // (identical across lanes of each half since B was column-broadcast).
    if (lane == 0) {
        #pragma unroll
        for (int rc = 0; rc < 4; ++rc)
            #pragma unroll
            for (int t = 0; t < 8; ++t) gbuf[rc * 16 + t] = acc[rc][t];
    } else if (lane == 16) {
        #pragma unroll
        for (int rc = 0; rc < 4; ++rc)
            #pragma unroll
            for (int t = 0; t < 8; ++t) gbuf[rc * 16 + 8 + t] = acc[rc][t];
    }
    __syncthreads();

    if (lane < 16) {
        const size_t bb = (size_t)e * 64 + lane;
        float gi = gbuf[lane]      + nt_load1(&b_ih[bb])      + nt_load1(&b_hh[bb]);
        float gf = gbuf[16 + lane] + nt_load1(&b_ih[bb + 16]) + nt_load1(&b_hh[bb + 16]);
        float gg = gbuf[32 + lane] + nt_load1(&b_ih[bb + 32]) + nt_load1(&b_hh[bb + 32]);
        float go = gbuf[48 + lane] + nt_load1(&b_ih[bb + 48]) + nt_load1(&b_hh[bb + 48]);
        float c  = sigmoidf_(gf) * c0[lane] + sigmoidf_(gi) * tanhf(gg);
        float m  = sigmoidf_(go) * tanhf(c);
        atomicAdd(&agg[(size_t)tgt * HDIM + lane], m);
    }
}

// ---------------------------------------------------------------------------
// Node kernel: one wave per node.
// 1) local LSTM (same WMMA scheme),
// 2) z2 = concat(local, agg[n]),
// 3) h1 = fin_W1[n](128x32) @ z2 via WMMA, LayerNorm + ReLU + dot(fin_W2).
// ---------------------------------------------------------------------------
__global__ __launch_bounds__(256) void UMPNN_node_kernel(
    const float* __restrict__ x,
    const float* __restrict__ h0, const float* __restrict__ c0,
    const float* __restrict__ uW_ih, const float* __restrict__ uW_hh,
    const float* __restrict__ ub_ih, const float* __restrict__ ub_hh,
    const float* __restrict__ W1, const float* __restrict__ ln_gamma,
    const float* __restrict__ ln_beta, const float* __restrict__ W2,
    const float* __restrict__ agg, float* __restrict__ out)
{
    __shared__ __align__(16) float lds[8 * 224];  // per wave: 32 z + 64 g + 128 h1
    const int lane = threadIdx.x & 31;
    const int wave = threadIdx.x >> 5;
    const int n = blockIdx.x * 8 + wave;

    float* zbuf = &lds[wave * 224];
    float* gbuf = zbuf + 32;
    float* hbuf = gbuf + 64;

    if (lane < 16) zbuf[lane] = x[(size_t)n * IDIM + lane];
    else           zbuf[lane] = h0[lane - 16];
    __syncthreads();

    const float* Wih = uW_ih + (size_t)n * 64 * 16;
    const float* Whh = uW_hh + (size_t)n * 64 * 16;
    const int row = lane & 15;
    const int h2  = (lane >> 4) << 1;

    v8f acc[4];
    #pragma unroll
    for (int rc = 0; rc < 4; ++rc) acc[rc] = (v8f){0,0,0,0,0,0,0,0};

    #pragma unroll
    for (int kc = 0; kc < 8; ++kc) {
        const int kk = kc * 4 + h2;
        v2f b = *(const v2f*)&zbuf[kk];
        const float* Wbase = (kk < 16) ? (Wih + kk) : (Whh + (kk - 16));
        #pragma unroll
        for (int rc = 0; rc < 4; ++rc) {
            v2f a = nt_load2(Wbase + (rc * 16 + row) * 16);
            acc[rc] = __builtin_amdgcn_wmma_f32_16x16x4_f32(
                false, a, false, b, (short)0, acc[rc], false, false);
        }
    }

    if (lane == 0) {
        #pragma unroll
        for (int rc = 0; rc < 4; ++rc)
            #pragma unroll
            for (int t = 0; t < 8; ++t) gbuf[rc * 16 + t] = acc[rc][t];
    } else if (lane == 16) {
        #pragma unroll
        for (int rc = 0; rc < 4; ++rc)
            #pragma unroll
            for (int t = 0; t < 8; ++t) gbuf[rc * 16 + 8 + t] = acc[rc][t];
    }
    __syncthreads();

    float loc = 0.0f;
    if (lane < 16) {
        const size_t bb = (size_t)n * 64 + lane;
        float gi = gbuf[lane]      + nt_load1(&ub_ih[bb])      + nt_load1(&ub_hh[bb]);
        float gf = gbuf[16 + lane] + nt_load1(&ub_ih[bb + 16]) + nt_load1(&ub_hh[bb + 16]);
        float gg = gbuf[32 + lane] + nt_load1(&ub_ih[bb + 32]) + nt_load1(&ub_hh[bb + 32]);
        float go = gbuf[48 + lane] + nt_load1(&ub_ih[bb + 48]) + nt_load1(&ub_hh[bb + 48]);
        float c  = sigmoidf_(gf) * c0[lane] + sigmoidf_(gi) * tanhf(gg);
        loc      = sigmoidf_(go) * tanhf(c);
    }
    // z2 = concat(local, aggregated messages)
    if (lane < 16) zbuf[lane] = loc;
    else           zbuf[lane] = agg[(size_t)n * HDIM + (lane - 16)];
    __syncthreads();

    // h1 = fin_W1[n] (128x32) @ z2 : 8 row-chunks x 8 k-chunks of WMMA
    const float* W1n = W1 + (size_t)n * KDIM * 32;
    v8f acc2[8];
    #pragma unroll
    for (int rc = 0; rc < 8; ++rc) acc2[rc] = (v8f){0,0,0,0,0,0,0,0};

    #pragma unroll
    for (int kc = 0; kc < 8; ++kc) {
        const int kk = kc * 4 + h2;
        v2f b = *(const v2f*)&zbuf[kk];
        #pragma unroll
        for (int rc = 0; rc < 8; ++rc) {
            v2f a = nt_load2(W1n + (size_t)(rc * 16 + row) * 32 + kk);
            acc2[rc] = __builtin_amdgcn_wmma_f32_16x16x4_f32(
                false, a, false, b, (short)0, acc2[rc], false, false);
        }
    }

    if (lane == 0) {
        #pragma unroll
        for (int rc = 0; rc < 8; ++rc)
            #pragma unroll
            for (int t = 0; t < 8; ++t) hbuf[rc * 16 + t] = acc2[rc][t];
    } else if (lane == 16) {
        #pragma unroll
        for (int rc = 0; rc < 8; ++rc)
            #pragma unroll
            for (int t = 0; t < 8; ++t) hbuf[rc * 16 + 8 + t] = acc2[rc][t];
    }
    __syncthreads();

    // LayerNorm(128) + ReLU + dot(W2): 4 elements per lane
    float hv[4], s = 0.0f, ss = 0.0f;
    #pragma unroll
    for (int j = 0; j < 4; ++j) {
        int k = lane + 32 * j;
        hv[j] = hbuf[k];
        s  += hv[j];
        ss += hv[j] * hv[j];
    }
    #pragma unroll
    for (int off = 16; off >= 1; off >>= 1) {
        s  += __shfl_xor(s,  off, 32);
        ss += __shfl_xor(ss, off, 32);
    }
    const float mu  = s * (1.0f / 128.0f);
    const float var = ss * (1.0f / 128.0f) - mu * mu;
    const float rs  = rsqrtf(var + LN_EPS);

    float part = 0.0f;
    #pragma unroll
    for (int j = 0; j < 4; ++j) {
        int k = lane + 32 * j;
        size_t idx = (size_t)n * KDIM + k;
        float a = (hv[j] - mu) * rs * nt_load1(&ln_gamma[idx]) + nt_load1(&ln_beta[idx]);
        a = fmaxf(a, 0.0f);
        part += nt_load1(&W2[idx]) * a;
    }
    #pragma unroll
    for (int off = 16; off >= 1; off >>= 1)
        part += __shfl_xor(part, off, 32);

    if (lane == 0) out[n] = part;
}

extern "C" void kernel_launch(void* const* d_in, const int* in_sizes, int n_in,
                              void* d_out, int out_size, void* d_ws, size_t ws_size,
                              hipStream_t stream) {
    const float* x        = (const float*)d_in[0];
    const int*   ei       = (const int*)  d_in[1];
    const float* h0       = (const float*)d_in[2];
    const float* c0       = (const float*)d_in[3];
    const float* msg_W_ih = (const float*)d_in[4];
    const float* msg_W_hh = (const float*)d_in[5];
    const float* msg_b_ih = (const float*)d_in[6];
    const float* msg_b_hh = (const float*)d_in[7];
    const float* upd_W_ih = (const float*)d_in[8];
    const float* upd_W_hh = (const float*)d_in[9];
    const float* upd_b_ih = (const float*)d_in[10];
    const float* upd_b_hh = (const float*)d_in[11];
    const float* fin_W1   = (const float*)d_in[12];
    const float* ln_gamma = (const float*)d_in[13];
    const float* ln_beta  = (const float*)d_in[14];
    const float* fin_W2   = (const float*)d_in[15];

    float* agg = (float*)d_ws;                      // [N, H] scatter buffer
    float* out = (float*)d_out;                     // [N]

    const int agg_elems = NNODES * HDIM;
    UMPNN_zero_kernel<<<(agg_elems + 255) / 256, 256, 0, stream>>>(agg, agg_elems);

    UMPNN_edge_kernel<<<NEDGES / 8, 256, 0, stream>>>(
        x, ei, h0, c0, msg_W_ih, msg_W_hh, msg_b_ih, msg_b_hh, agg);

    UMPNN_node_kernel<<<NNODES / 8, 256, 0, stream>>>(
        x, h0, c0, upd_W_ih, upd_W_hh, upd_b_ih, upd_b_hh,
        fin_W1, ln_gamma, ln_beta, fin_W2, agg, out);
}